// BilateralFilter_41678362640728
// MI455X (gfx1250) — compile-verified
//
#include <hip/hip_runtime.h>
#include <math.h>

// ---------------------------------------------------------------------------
// Per-batch Gaussian blur (k in {5..9}, separable), fused with the even-k
// bilinear resize.  Vertical 1-D pass runs on the CDNA5 matrix core as
// Out(16x16) = A_toeplitz(16x24) x H(24x16) via 6x V_WMMA_F32_16X16X4_F32.
// Memory-bound (~4.5 flop/byte << roofline at 23.3 TB/s), so one fused pass
// (read x once + halo, write out once).  Toeplitz A elements come from a
// zero-padded shifted weight table so the WMMA loop is entirely branchless.
// Output stores are non-temporal (write-once stream); input loads stay RT so
// halo re-reads between adjacent tiles hit the 192 MB L2.
// ---------------------------------------------------------------------------

typedef float v2f __attribute__((ext_vector_type(2)));
typedef float v8f __attribute__((ext_vector_type(8)));

#define BATCH 8
#define NCH   64
#define HW    384
#define MAXK  9

// per-wave LDS region (floats):
//   X  : 24 rows x stride 26 (24x24 input tile + halo, padded stride) = 624
//   H  : 24 x 16 horizontally-blurred tile                            = 384
//   S  : 16 x 17 conv-output staging for bilinear (even-k path)       = 272
//   We : 40-entry shifted weight table, We[15+t] = w[t], 0 elsewhere  =  40
#define XSTR   26
#define OFF_X  0
#define OFF_H  624
#define OFF_S  (624 + 384)
#define OFF_W  (624 + 384 + 272)
#define REG_SZ (624 + 384 + 272 + 40)   // 1320 floats / wave (x8 = 42.2 KB)

__device__ __forceinline__ float sigmoidf_(float v) {
    return 1.0f / (1.0f + expf(-v));
}

// ---- tiny per-batch param kernel: k_sel, sigma, normalized 1-D weights ----
__global__ void gauss_params_kernel(const float* __restrict__ params,
                                    int* __restrict__ hk,
                                    float* __restrict__ hw) {
    int b = threadIdx.x;
    if (b >= BATCH) return;
    float p0 = params[2 * b + 0];
    float p1 = params[2 * b + 1];
    float ki = truncf(p0);                                   // torch .long() == trunc
    int   k  = (int)floorf(5.0f + 5.0f * sigmoidf_(ki));     // in {5..9}
    if (k < 5) k = 5;
    if (k > 9) k = 9;
    float sig = 0.5f + 4.5f * sigmoidf_(p1);

    float w[12];
    for (int i = 0; i < 12; ++i) w[i] = 0.0f;                // taps >= k stay 0
    float sum = 0.0f;
    int   c   = k / 2;
    for (int i = 0; i < k; ++i) {
        float d = (float)(i - c);
        float v = expf(-(d * d) / (2.0f * sig * sig));
        w[i] = v;
        sum += v;
    }
    for (int i = 0; i < k; ++i) w[i] /= sum;   // 1-D norm; 2-D kernel = outer product

    hk[b] = k;
    for (int i = 0; i < 12; ++i) hw[b * 12 + i] = w[i];
}

// ---- fused separable blur (+ bilinear resize for even k) ----
__launch_bounds__(256)
__global__ void gauss_blur_kernel(const float* __restrict__ x,
                                  const int* __restrict__ hk,
                                  const float* __restrict__ hw,
                                  float* __restrict__ out) {
    __shared__ float lds[8 * REG_SZ];

    const int lane = threadIdx.x & 31;
    const int wave = threadIdx.x >> 5;
    float* Xs = &lds[wave * REG_SZ + OFF_X];
    float* Hs = &lds[wave * REG_SZ + OFF_H];
    float* Ss = &lds[wave * REG_SZ + OFF_S];
    float* We = &lds[wave * REG_SZ + OFF_W];

    const int img = blockIdx.z;          // b*64 + ch
    const int b   = img >> 6;
    const int k   = hk[b];
    const int pad = k >> 1;
    const bool even = (k & 1) == 0;

    // odd k: conv out 384x384 -> 24x24 tiles of 16 (exact cover, no edge guard)
    // even k: conv out 385x385, final 384x384 -> 28x28 tiles of 14 final px,
    //         each needing a 16x16 conv region (13*385/384 + 2 < 16).
    const int tpr    = even ? 28 : 24;
    const int ntiles = tpr * tpr;
    const int t      = blockIdx.x * 8 + wave;   // one tile per wave (uniform branch)
    if (t >= ntiles) return;
    const int ty = t / tpr, tx = t - ty * tpr;

    // shifted, zero-padded weight table: We[15 + t] = w[t] (w[t]=0 for t>=k)
    for (int i = lane; i < 40; i += 32) {
        int tt = i - 15;
        We[i] = (tt >= 0 && tt < 12) ? hw[b * 12 + tt] : 0.0f;
    }

    const float scale = 385.0f / 384.0f;        // half-pixel resize ratio
    int jr0, jc0, fr0 = 0, fc0 = 0;
    if (even) {
        fr0 = ty * 14; fc0 = tx * 14;
        jr0 = (int)floorf((fr0 + 0.5f) * scale - 0.5f);
        jc0 = (int)floorf((fc0 + 0.5f) * scale - 0.5f);
    } else {
        jr0 = ty * 16; jc0 = tx * 16;
    }
    const int ir0 = jr0 - pad, ic0 = jc0 - pad;
    const float* xin = x + (size_t)img * (HW * HW);

    // ---- stage 24x24 input tile (zero padding outside the image) ----
    for (int i = lane; i < 576; i += 32) {
        int j = i / 24, cc = i - j * 24;
        int gr = ir0 + j, gc = ic0 + cc;
        float v = 0.0f;
        if (gr >= 0 && gr < HW && gc >= 0 && gc < HW) v = xin[gr * HW + gc];
        Xs[j * XSTR + cc] = v;
    }
    asm volatile("s_wait_dscnt 0" ::: "memory");   // same-wave DS in order

    // ---- horizontal 1-D pass (VALU), branchless: taps >= k are 0 ----
    for (int i = lane; i < 384; i += 32) {
        int j = i >> 4, c = i & 15;
        float s = 0.0f;
#pragma unroll
        for (int tt = 0; tt < MAXK; ++tt)
            s += We[15 + tt] * Xs[j * XSTR + c + tt];
        Hs[j * 16 + c] = s;
    }
    asm volatile("s_wait_dscnt 0" ::: "memory");

    // ---- vertical 1-D pass as WMMA: Out = Toeplitz(w) x H ----
    // f32 16x16x4 layouts (ISA 7.12.2):
    //   A: lane l -> M=l&15, K = 4s + 2*(l>>4) + {0,1}
    //   B: lane l -> N=l&15, K rows = 4s + 2*(l>>4) + {0,1}
    //   D: VGPR v -> M = v + 8*(l>>4), N = l&15
    v8f acc = {0.f, 0.f, 0.f, 0.f, 0.f, 0.f, 0.f, 0.f};
    const int m     = lane & 15;
    const int half  = lane >> 4;
    const int wbase = 15 - m;                    // We index base for this lane's row
#pragma unroll
    for (int s4 = 0; s4 < 6; ++s4) {
        const int kk0 = s4 * 4 + half * 2;
        v2f a, bb;
        a.x = We[wbase + kk0];                   // A[r][j] = w[j-r], zero-padded
        a.y = We[wbase + kk0 + 1];
        bb.x = Hs[kk0 * 16 + m];
        bb.y = Hs[(kk0 + 1) * 16 + m];
        acc = __builtin_amdgcn_wmma_f32_16x16x4_f32(
            /*neg_a=*/false, a, /*neg_b=*/false, bb,
            /*c_mod=*/(short)0, acc, /*reuse_a=*/false, /*reuse_b=*/false);
    }

    float* yout = out + (size_t)img * (HW * HW);

    if (!even) {
        // direct store of the 16x16 conv tile; 24x16 tiling covers 384 exactly,
        // so no bounds check is needed (rr,cc <= 383 by construction).
        const int rrb = jr0 + half * 8;
        const int cc  = jc0 + m;
#pragma unroll
        for (int v = 0; v < 8; ++v)
            __builtin_nontemporal_store(acc[v], &yout[(rrb + v) * HW + cc]);
    } else {
        // stage conv tile, then fused half-pixel bilinear 385->384
#pragma unroll
        for (int v = 0; v < 8; ++v)
            Ss[(v + half * 8) * 17 + m] = acc[v];
        asm volatile("s_wait_dscnt 0" ::: "memory");

        for (int i = lane; i < 196; i += 32) {         // 14x14 final tile
            int dr = i / 14, dc = i - dr * 14;
            int fr = fr0 + dr, fc = fc0 + dc;
            if (fr >= HW || fc >= HW) continue;
            float sy = (fr + 0.5f) * scale - 0.5f;     // >= 0 for fr>=0
            float sx = (fc + 0.5f) * scale - 0.5f;
            int y0 = (int)floorf(sy), x0 = (int)floorf(sx);
            float wy = sy - (float)y0, wx = sx - (float)x0;
            int ly0 = y0 - jr0, lx0 = x0 - jc0;        // in [0,15]
            int ly1 = (y0 + 1 > 384) ? ly0 : ly0 + 1;  // clamp at conv edge (Ho-1=384)
            int lx1 = (x0 + 1 > 384) ? lx0 : lx0 + 1;
            float v00 = Ss[ly0 * 17 + lx0], v01 = Ss[ly0 * 17 + lx1];
            float v10 = Ss[ly1 * 17 + lx0], v11 = Ss[ly1 * 17 + lx1];
            float val = (1.0f - wy) * ((1.0f - wx) * v00 + wx * v01)
                      +         wy * ((1.0f - wx) * v10 + wx * v11);
            __builtin_nontemporal_store(val, &yout[fr * HW + fc]);
        }
    }
}

extern "C" void kernel_launch(void* const* d_in, const int* in_sizes, int n_in,
                              void* d_out, int out_size, void* d_ws, size_t ws_size,
                              hipStream_t stream) {
    (void)in_sizes; (void)n_in; (void)out_size; (void)ws_size;
    const float* x      = (const float*)d_in[0];
    const float* params = (const float*)d_in[1];
    float* out = (float*)d_out;

    int*   hk = (int*)d_ws;                 // 8 ints
    float* hw = (float*)d_ws + 16;          // 8 x 12 floats @ byte offset 64

    gauss_params_kernel<<<1, 32, 0, stream>>>(params, hk, hw);

    // worst case tiles/image: 28*28 = 784; 8 waves (tiles) per 256-thread block
    dim3 grid((784 + 7) / 8, 1, BATCH * NCH);
    gauss_blur_kernel<<<grid, 256, 0, stream>>>(x, hk, hw, out);
}